// CombinedGraphLoss_87660282511506
// MI455X (gfx1250) — compile-verified
//
#include <hip/hip_runtime.h>
#include <hip/hip_bf16.h>

#define N 8192
#define ALPHA 0.1f
#define LAMBDA_SMOOTH 0.1f
#define GAMMA 0.001f

typedef __attribute__((ext_vector_type(2))) float v2f;
typedef __attribute__((ext_vector_type(8))) float v8f;

// -----------------------------------------------------------------------------
// Kernel 1: d[i] = sum_j relu(adj[i][j]) via V_WMMA_F32_16X16X4_F32 with B = ones.
// Coalesced b128 global loads stage a 16x256 tile in LDS (rows padded to 260
// floats to avoid bank conflicts on the column-strided WMMA A fetch).
// Each of the 8 waves owns a 32-column slice -> 8 WMMA steps (K=4) per tile.
// D's 16 columns are all identical row-sums; lanes 0 and 16 hold M=0..7 / 8..15.
// -----------------------------------------------------------------------------
__global__ __launch_bounds__(256) void rowsum_dinv_kernel(
    const float* __restrict__ adj, float* __restrict__ dinv)
{
    __shared__ float tile[16][260];   // 256 cols + 4 pad
    __shared__ float wpart[8][16];

    const int t    = threadIdx.x;
    const int wave = t >> 5;
    const int lane = t & 31;
    const long long rowbase = (long long)blockIdx.x * 16;

    v8f c = {};                        // f32 16x16 accumulator (8 VGPRs)
    const v2f bones = {1.0f, 1.0f};    // all-ones B (layout irrelevant)

    const int myrow = lane & 15;       // M mapping: lanes 0-15 and 16-31 -> M=0..15
    const int khalf = (lane >> 4) * 2; // lanes 16-31 supply K=2,3 of each K=4 step

    for (int tilei = 0; tilei < N / 256; ++tilei) {
        const int c0 = tilei * 256;
        // cooperative coalesced load + relu of the 16x256 tile
        #pragma unroll
        for (int q = 0; q < 4; ++q) {
            const int f  = t + q * 256;     // float4 index within tile
            const int r  = f >> 6;          // 64 float4s per row
            const int c4 = f & 63;
            float4 v = *(const float4*)(adj + (rowbase + r) * (long long)N + c0 + c4 * 4);
            v.x = fmaxf(v.x, 0.0f); v.y = fmaxf(v.y, 0.0f);
            v.z = fmaxf(v.z, 0.0f); v.w = fmaxf(v.w, 0.0f);
            *(float4*)&tile[r][c4 * 4] = v;
        }
        __syncthreads();

        #pragma unroll
        for (int s = 0; s < 8; ++s) {
            const int kbase = wave * 32 + s * 4 + khalf;
            v2f a;
            a.x = tile[myrow][kbase];
            a.y = tile[myrow][kbase + 1];
            // (neg_a, A, neg_b, B, c_mod, C, reuse_a, reuse_b)
            c = __builtin_amdgcn_wmma_f32_16x16x4_f32(
                    false, a, false, bones, (short)0, c, false, false);
        }
        __syncthreads();
    }

    // Extract row sums: VGPR r of lane 0 = row r, of lane 16 = row 8+r.
    if (lane == 0) {
        #pragma unroll
        for (int r = 0; r < 8; ++r) wpart[wave][r] = c[r];
    } else if (lane == 16) {
        #pragma unroll
        for (int r = 0; r < 8; ++r) wpart[wave][8 + r] = c[r];
    }
    __syncthreads();

    if (t < 16) {
        float s = 0.0f;
        #pragma unroll
        for (int w = 0; w < 8; ++w) s += wpart[w][t];
        dinv[rowbase + t] = 1.0f / sqrtf(s + 1e-10f);
    }
}

// -----------------------------------------------------------------------------
// Kernel 2: one block per row i; thread t covers columns t + 256k, k = 0..31.
// A[i][j] = dinv[i] * relu(adj[i][j]) * dinv[j]
// acc += (A - exp(-a*|i-j|))^2 + GAMMA*A^2 + (interior) LAMBDA*|A - 0.25*stencil|
// Neighbor-row reads hit L2 (re-read by adjacent blocks). Block partial -> ws.
// -----------------------------------------------------------------------------
__global__ __launch_bounds__(256) void loss_kernel(
    const float* __restrict__ adj, const float* __restrict__ dinv,
    float* __restrict__ partial)
{
    const int i = blockIdx.x;
    const int t = threadIdx.x;
    const float di   = dinv[i];
    const float dim1 = (i > 0)     ? dinv[i - 1] : 0.0f;
    const float dip1 = (i < N - 1) ? dinv[i + 1] : 0.0f;
    const long long rowoff = (long long)i * N;
    const bool irow = (i >= 1) && (i <= N - 2);

    float acc = 0.0f;
    for (int k = 0; k < 32; ++k) {
        const int j = t + k * 256;
        const float aij = fmaxf(adj[rowoff + j], 0.0f);
        const float dj  = dinv[j];
        const float cen = di * aij * dj;
        const float dist = fabsf((float)(i - j));
        const float dec  = __expf(-ALPHA * dist);
        float term = (cen - dec) * (cen - dec) + GAMMA * cen * cen;
        if (irow && j >= 1 && j <= N - 2) {
            const float up = dim1 * fmaxf(adj[rowoff - N + j], 0.0f) * dj;
            const float dn = dip1 * fmaxf(adj[rowoff + N + j], 0.0f) * dj;
            const float lf = di * fmaxf(adj[rowoff + j - 1], 0.0f) * dinv[j - 1];
            const float rt = di * fmaxf(adj[rowoff + j + 1], 0.0f) * dinv[j + 1];
            term += LAMBDA_SMOOTH * fabsf(cen - 0.25f * (up + dn + lf + rt));
        }
        acc += term;
    }

    __shared__ float red[256];
    red[t] = acc;
    __syncthreads();
    for (int s = 128; s > 0; s >>= 1) {
        if (t < s) red[t] += red[t + s];
        __syncthreads();
    }
    if (t == 0) partial[i] = red[0];
}

// -----------------------------------------------------------------------------
// Kernel 3: reduce 8192 row partials -> d_out[0] (direct store, no atomics).
// -----------------------------------------------------------------------------
__global__ __launch_bounds__(256) void final_reduce_kernel(
    const float* __restrict__ partial, float* __restrict__ out)
{
    const int t = threadIdx.x;
    float acc = 0.0f;
    for (int k = t; k < N; k += 256) acc += partial[k];
    __shared__ float red[256];
    red[t] = acc;
    __syncthreads();
    for (int s = 128; s > 0; s >>= 1) {
        if (t < s) red[t] += red[t + s];
        __syncthreads();
    }
    if (t == 0) out[0] = red[0];
}

extern "C" void kernel_launch(void* const* d_in, const int* in_sizes, int n_in,
                              void* d_out, int out_size, void* d_ws, size_t ws_size,
                              hipStream_t stream)
{
    const float* adj = (const float*)d_in[0];
    float* ws      = (float*)d_ws;
    float* dinv    = ws;        // N floats
    float* partial = ws + N;    // N floats
    float* out     = (float*)d_out;

    rowsum_dinv_kernel<<<N / 16, 256, 0, stream>>>(adj, dinv);
    loss_kernel<<<N, 256, 0, stream>>>(adj, dinv, partial);
    final_reduce_kernel<<<1, 256, 0, stream>>>(partial, out);
}